// FilterDetections_29978871726712
// MI455X (gfx1250) — compile-verified
//
#include <hip/hip_runtime.h>
#include <math.h>

#ifndef __has_builtin
#define __has_builtin(x) 0
#endif

// ---------------- problem constants (match reference) ----------------
#define NN      196416
#define CC      8
#define MAXDET  100
#define TPB     256            // 8 waves (wave32)
#define EPT     12             // scores per thread, VGPR-resident
#define TILE    (TPB * EPT)    // 3072 boxes per block (48KB LDS tile)
#define NBC     ((NN + TILE - 1) / TILE)   // 64 blocks per class
#define NEG_INF (-INFINITY)

// workspace word offsets (4-byte words)
#define WS_BAR   0      // 8 classes * 32-word stride  (256 words)
#define WS_PVAL  256    // C*NBC floats                (512)
#define WS_PIDX  768    // C*NBC ints                  (512)
#define WS_WIN   1280   // C*8 floats (box4, keep)     (64)
#define WS_DIDX  1344   // C*MAXDET ints               (800)
#define WS_CAND  2144   // C*MAXDET floats             (800)  -- contiguous after WS_DIDX

typedef unsigned int v4u __attribute__((ext_vector_type(4)));
typedef int          v4i __attribute__((ext_vector_type(4)));
typedef int          v8i __attribute__((ext_vector_type(8)));

#if __has_builtin(__builtin_amdgcn_tensor_load_to_lds)
#define HAS_TDM 1
#else
#define HAS_TDM 0
#endif

// TDM availability must hold in the *device* pass only — the host pass of the
// HIP TU targets x86 where the AMDGCN builtin is (correctly) absent.
#if defined(__HIP_DEVICE_COMPILE__)
static_assert(HAS_TDM == 1, "gfx1250 tensor_load_to_lds builtin not available");
#endif

#if HAS_TDM
// 1D bulk DMA global->LDS via Tensor Data Mover. Degenerate 2D descriptor:
// data_size=4B, one row of `tileElems` elements; elements past `tensorElems`
// read as zero (used to zero-fill the ragged last tile).
__device__ __forceinline__ void tdm_load_1d(unsigned ldsAddr, const void* gsrc,
                                            unsigned tileElems,
                                            unsigned tensorElems) {
  unsigned long long ga = (unsigned long long)(uintptr_t)gsrc;
  v4u g0;
  g0.x = 1u;                                  // count=1 (valid descriptor)
  g0.y = ldsAddr;                             // lds_addr (bytes)
  g0.z = (unsigned)ga;                        // global_addr[31:0]
  g0.w = (unsigned)(ga >> 32) | (2u << 30);   // global_addr[56:32] | type=2
  v8i g1 = { (int)(2u << 16),                       // data_size = 4B
             (int)(tensorElems << 16),               // tensor_dim0[15:0]
             (int)((tensorElems >> 16) | (1u << 16)),// tensor_dim0 hi | tensor_dim1=1
             (int)(tileElems << 16),                 // tile_dim0
             1,                                      // tile_dim1=1, tile_dim2=0
             (int)tensorElems,                       // tensor_dim0_stride lo
             0, 0 };
  v4i gz = { 0, 0, 0, 0 };
#if defined(__clang_major__) && (__clang_major__ >= 23)
  v8i gz8 = { 0, 0, 0, 0, 0, 0, 0, 0 };
  __builtin_amdgcn_tensor_load_to_lds(g0, g1, gz, gz, gz8, 0);
#else
  __builtin_amdgcn_tensor_load_to_lds(g0, g1, gz, gz, 0);
#endif
}

__device__ __forceinline__ void tdm_wait() {
#if __has_builtin(__builtin_amdgcn_s_wait_tensorcnt)
  __builtin_amdgcn_s_wait_tensorcnt(0);
#else
  asm volatile("s_wait_tensorcnt 0x0" ::: "memory");
#endif
}
#endif

// monotonic-counter per-class grid barrier (counters reset each launch)
__device__ __forceinline__ void class_sync(unsigned* bar, unsigned need) {
  __syncthreads();
  if (threadIdx.x == 0) {
    __threadfence();
    atomicAdd(bar, 1u);
    while (atomicAdd(bar, 0u) < need) __builtin_amdgcn_s_sleep(2);
    __threadfence();
  }
  __syncthreads();
}

__global__ void init_kernel(unsigned* wsu) {
  if (threadIdx.x < 256) wsu[WS_BAR + threadIdx.x] = 0u;
}

// Persistent per-class NMS: grid = (NBC, CC). Scores in VGPRs, boxes in LDS
// (loaded once via TDM DMA), 100 rounds with 2 per-class global syncs each.
__global__ void __launch_bounds__(TPB)
nms_kernel(const float* __restrict__ boxes, const float* __restrict__ clsf,
           float* wsf) {
  int*      wsi = (int*)wsf;
  unsigned* wsu = (unsigned*)wsf;

  const int c        = blockIdx.y;
  const int blk      = blockIdx.x;
  const int t        = threadIdx.x;
  const int tileBase = blk * TILE;

  const float* bx3 = boxes + (size_t)3 * NN * 4;   // last batch element only
  const float* cl3 = clsf  + (size_t)3 * NN * CC;

  __shared__ float4 sBoxes[TILE];   // 48KB box tile, reused for all 100 rounds
  __shared__ float  sval[TPB];
  __shared__ int    sidx[TPB];
  __shared__ float  sWin[8];

  // ---- stage boxes tile into LDS via TDM DMA; overlap with score loads ----
#if HAS_TDM
  if (t == 0) {   // exactly one wave issues the DMA (EXEC==0 waves branch over)
    tdm_load_1d((unsigned)(uintptr_t)(&sBoxes[0]),
                bx3 + (size_t)tileBase * 4,
                (unsigned)(TILE * 4),                 // 12288 f32 elements
                (unsigned)((NN - tileBase) * 4));     // OOB -> zero boxes
  }
#else
  for (int e = t; e < TILE; e += TPB) {
    int g = tileBase + e;
    float4 v = make_float4(0.f, 0.f, 0.f, 0.f);
    if (g < NN) v = ((const float4*)bx3)[g];
    sBoxes[e] = v;
  }
#endif

  // per-thread thresholded scores (VGPR resident) — overlaps the TDM DMA
  float sc[EPT];
  const int myBase = tileBase + t * EPT;
#pragma unroll
  for (int j = 0; j < EPT; ++j) {
    int g = myBase + j;
    float v = NEG_INF;
    if (g < NN) {
      float s = cl3[(size_t)g * CC + c];
      v = (s > 0.01f) ? s : NEG_INF;             // SCORE_THRESH
    }
    sc[j] = v;
  }

#if HAS_TDM
  if (t == 0) tdm_wait();
#endif
  __syncthreads();

  unsigned* bar  = wsu + WS_BAR  + c * 32;
  float*    pval = wsf + WS_PVAL + c * NBC;
  int*      pidx = wsi + WS_PIDX + c * NBC;
  float*    win  = wsf + WS_WIN  + c * 8;
  int*      didx = wsi + WS_DIDX + c * MAXDET;
  float*    cand = wsf + WS_CAND + c * MAXDET;

  unsigned syncs = 0;

  for (int r = 0; r < MAXDET; ++r) {
    // ---- local argmax (first-occurrence tie-break, like jnp.argmax) ----
    float bv = sc[0]; int bj = 0;
#pragma unroll
    for (int j = 1; j < EPT; ++j)
      if (sc[j] > bv) { bv = sc[j]; bj = j; }
    sval[t] = bv; sidx[t] = myBase + bj;
    __syncthreads();
    for (int s = TPB >> 1; s > 0; s >>= 1) {
      if (t < s) {
        float ov = sval[t + s]; int oi = sidx[t + s];
        if (ov > sval[t] || (ov == sval[t] && oi < sidx[t])) {
          sval[t] = ov; sidx[t] = oi;
        }
      }
      __syncthreads();
    }
    if (t == 0) { pval[blk] = sval[0]; pidx[blk] = sidx[0]; }
    ++syncs; class_sync(bar, syncs * NBC);

    // ---- class leader: global argmax, record detection, publish winner ----
    if (blk == 0 && t == 0) {
      float gv = pval[0]; int gi = pidx[0];
      for (int b2 = 1; b2 < NBC; ++b2) {
        float ov = pval[b2]; int oi = pidx[b2];
        if (ov > gv || (ov == gv && oi < gi)) { gv = ov; gi = oi; }
      }
      int keep = (gv > NEG_INF) ? 1 : 0;
      float4 wb = ((const float4*)bx3)[gi];
      win[0] = wb.x; win[1] = wb.y; win[2] = wb.z; win[3] = wb.w;
      win[4] = (float)keep;
      didx[r] = gi;
      cand[r] = keep ? gv : NEG_INF;   // == cl[gi,c] when kept
    }
    ++syncs; class_sync(bar, syncs * NBC);

    // ---- suppression (skipped entirely when keep==0, matching reference) ----
    if (t == 0) {
      sWin[0] = win[0]; sWin[1] = win[1]; sWin[2] = win[2];
      sWin[3] = win[3]; sWin[4] = win[4];
    }
    __syncthreads();
    if (sWin[4] != 0.0f) {
      float wx1 = sWin[0], wy1 = sWin[1], wx2 = sWin[2], wy2 = sWin[3];
      float areaW = (wx2 - wx1) * (wy2 - wy1);
#pragma unroll
      for (int j = 0; j < EPT; ++j) {
        float4 b = sBoxes[t * EPT + j];
        float ix1 = fmaxf(wx1, b.x), iy1 = fmaxf(wy1, b.y);
        float ix2 = fminf(wx2, b.z), iy2 = fminf(wy2, b.w);
        float inter = fmaxf(ix2 - ix1, 0.f) * fmaxf(iy2 - iy1, 0.f);
        float areaB = (b.z - b.x) * (b.w - b.y);
        float uni = areaW + areaB - inter;
        float iou = inter / fmaxf(uni, 1e-8f);
        if (iou > 0.5f) sc[j] = NEG_INF;   // winner self-suppresses (iou==1)
      }
    }
    __syncthreads();
  }
}

// Single block: stable top-100 over 800 candidates + final gather/pad.
// Candidate block (didx[800] | cand[800], contiguous in ws) is staged into
// LDS with a single TDM DMA.
__global__ void __launch_bounds__(TPB)
topk_kernel(const float* __restrict__ boxes, const float* __restrict__ rot,
            const float* __restrict__ trn, const float* __restrict__ hand,
            const float* wsf, float* __restrict__ out) {
  const int t     = threadIdx.x;
  const int TOTAL = CC * MAXDET;   // 800

  __shared__ int           scanbuf[2 * CC * MAXDET]; // [0..799]=didx, [800..1599]=cand
  __shared__ unsigned char used[CC * MAXDET];
  __shared__ float         rv[TPB];
  __shared__ int           ri[TPB];
  __shared__ float         tsc[MAXDET];
  __shared__ int           tfi[MAXDET];

  int*   sdix  = scanbuf;
  float* scand = (float*)(scanbuf + TOTAL);

#if HAS_TDM
  if (t == 0) {
    tdm_load_1d((unsigned)(uintptr_t)scanbuf, wsf + WS_DIDX,
                (unsigned)(2 * TOTAL), (unsigned)(2 * TOTAL));
  }
  for (int j = t; j < TOTAL; j += TPB) used[j] = 0;   // overlap with DMA
  if (t == 0) tdm_wait();
#else
  const int* wsi = (const int*)wsf;
  for (int j = t; j < TOTAL; j += TPB) {
    scand[j] = wsf[WS_CAND + j];
    sdix[j]  = wsi[WS_DIDX + j];
    used[j]  = 0;
  }
#endif
  __syncthreads();

  for (int k = 0; k < MAXDET; ++k) {
    float bv = 0.f; int bi = -1;
    for (int j = t; j < TOTAL; j += TPB) {
      if (!used[j]) {
        float v = scand[j];
        if (bi < 0 || v > bv || (v == bv && j < bi)) { bv = v; bi = j; }
      }
    }
    rv[t] = bv; ri[t] = bi;
    __syncthreads();
    for (int s = TPB >> 1; s > 0; s >>= 1) {
      if (t < s) {
        float ov = rv[t + s]; int oi = ri[t + s];
        if (oi >= 0 &&
            (ri[t] < 0 || ov > rv[t] || (ov == rv[t] && oi < ri[t]))) {
          rv[t] = ov; ri[t] = oi;
        }
      }
      __syncthreads();
    }
    if (t == 0) { tsc[k] = rv[0]; tfi[k] = ri[0]; used[ri[0]] = 1; }
    __syncthreads();
  }

  // outputs (flat f32, return order): boxes[400] scores[100] labels[100]
  //                                   rot[300] tr[300] hand[6300] = 7500
  const float* bx3 = boxes + (size_t)3 * NN * 4;
  const float* rt3 = rot   + (size_t)3 * NN * 3;
  const float* tr3 = trn   + (size_t)3 * NN * 3;
  const float* hd3 = hand  + (size_t)3 * NN * 63;

  for (int j = t; j < 7500; j += TPB) {
    float v; int k, d;
    if (j < 400) {
      k = j >> 2; d = j & 3;
      v = (tsc[k] > NEG_INF) ? bx3[(size_t)sdix[tfi[k]] * 4 + d] : -1.f;
    } else if (j < 500) {
      k = j - 400;
      v = (tsc[k] > NEG_INF) ? tsc[k] : -1.f;
    } else if (j < 600) {
      k = j - 500;
      v = (tsc[k] > NEG_INF) ? (float)(tfi[k] / MAXDET) : -1.f;
    } else if (j < 900) {
      k = (j - 600) / 3; d = (j - 600) % 3;
      v = (tsc[k] > NEG_INF) ? rt3[(size_t)sdix[tfi[k]] * 3 + d] : -1.f;
    } else if (j < 1200) {
      k = (j - 900) / 3; d = (j - 900) % 3;
      v = (tsc[k] > NEG_INF) ? tr3[(size_t)sdix[tfi[k]] * 3 + d] : -1.f;
    } else {
      k = (j - 1200) / 63; d = (j - 1200) % 63;
      v = (tsc[k] > NEG_INF) ? hd3[(size_t)sdix[tfi[k]] * 63 + d] : -1.f;
    }
    out[j] = v;
  }
}

extern "C" void kernel_launch(void* const* d_in, const int* in_sizes, int n_in,
                              void* d_out, int out_size, void* d_ws,
                              size_t ws_size, hipStream_t stream) {
  const float* boxes = (const float*)d_in[0];
  const float* clsf  = (const float*)d_in[1];
  const float* rot   = (const float*)d_in[2];
  const float* trn   = (const float*)d_in[3];
  const float* hand  = (const float*)d_in[4];
  float* wsf = (float*)d_ws;
  float* out = (float*)d_out;

  init_kernel<<<1, 256, 0, stream>>>((unsigned*)d_ws);
  nms_kernel<<<dim3(NBC, CC), TPB, 0, stream>>>(boxes, clsf, wsf);
  topk_kernel<<<1, TPB, 0, stream>>>(boxes, rot, trn, hand, wsf, out);
}